// FastWorkingMemory_74251394613693
// MI455X (gfx1250) — compile-verified
//
#include <hip/hip_runtime.h>
#include <math.h>

typedef __attribute__((ext_vector_type(2))) float v2f;
typedef __attribute__((ext_vector_type(8))) float v8f;

#define BB 4
#define TT 2048
#define DD 1024
#define HH 8
#define HD 128   // head dim

// ---------------------------------------------------------------------------
// GEMM: Y[n,o] = sum_d X[n,d] * W[o,d]   (i.e. Y = X @ W^T), all f32.
// Register-blocked: one wave computes a 32x32 output tile via 4 independent
// V_WMMA_F32_16X16X4_F32 accumulators, so each A/B fragment load feeds two
// WMMAs (4 loads : 4 WMMAs per K-step instead of 2:1).
// grid = (colTiles32/8, rowTiles32, nmat), block = 256 (8 waves).
// blockIdx.z selects among up to 4 (W, Y) pairs so all four projections run
// in a single launch.
// ---------------------------------------------------------------------------
__global__ __launch_bounds__(256) void gemm_xwT_f32(
    const float* __restrict__ X,
    const float* __restrict__ W0, const float* __restrict__ W1,
    const float* __restrict__ W2, const float* __restrict__ W3,
    float* __restrict__ Y0, float* __restrict__ Y1,
    float* __restrict__ Y2, float* __restrict__ Y3)
{
    const float* W;
    float* Y;
    switch (blockIdx.z) {
        case 0:  W = W0; Y = Y0; break;
        case 1:  W = W1; Y = Y1; break;
        case 2:  W = W2; Y = Y2; break;
        default: W = W3; Y = Y3; break;
    }

    const int wave = threadIdx.x >> 5;
    const int lane = threadIdx.x & 31;
    const int rowBase = blockIdx.y * 32;                  // 32 output rows
    const int colBase = (blockIdx.x * 8 + wave) * 32;     // 32 output cols

    // A fragment: lane L holds X[row + (L&15)][k + 2*(L>>4) + {0,1}]
    // B fragment: lane L holds W[col + (L&15)][k + 2*(L>>4) + {0,1}]
    const int rsel = lane & 15;
    const int koff = (lane >> 4) * 2;
    const float* __restrict__ xp0 = X + (size_t)(rowBase + rsel) * DD + koff;
    const float* __restrict__ xp1 = xp0 + (size_t)16 * DD;
    const float* __restrict__ wp0 = W + (size_t)(colBase + rsel) * DD + koff;
    const float* __restrict__ wp1 = wp0 + (size_t)16 * DD;

    v8f c00 = {}, c01 = {}, c10 = {}, c11 = {};
    #pragma unroll 4
    for (int k = 0; k < DD; k += 4) {
        v2f a0 = *(const v2f*)(xp0 + k);
        v2f a1 = *(const v2f*)(xp1 + k);
        v2f b0 = *(const v2f*)(wp0 + k);
        v2f b1 = *(const v2f*)(wp1 + k);
        c00 = __builtin_amdgcn_wmma_f32_16x16x4_f32(false, a0, false, b0,
                                                    (short)0, c00, false, false);
        c01 = __builtin_amdgcn_wmma_f32_16x16x4_f32(false, a0, false, b1,
                                                    (short)0, c01, false, false);
        c10 = __builtin_amdgcn_wmma_f32_16x16x4_f32(false, a1, false, b0,
                                                    (short)0, c10, false, false);
        c11 = __builtin_amdgcn_wmma_f32_16x16x4_f32(false, a1, false, b1,
                                                    (short)0, c11, false, false);
    }

    // D layout: VGPR r, lane L -> (M = r + 8*(L>>4), N = L&15)
    const int m0 = rowBase + 8 * (lane >> 4);
    const int n0 = colBase + (lane & 15);
    #pragma unroll
    for (int r = 0; r < 8; ++r) {
        Y[(size_t)(m0 + r) * DD + n0]            = c00[r];
        Y[(size_t)(m0 + r) * DD + n0 + 16]       = c01[r];
        Y[(size_t)(m0 + 16 + r) * DD + n0]       = c10[r];
        Y[(size_t)(m0 + 16 + r) * DD + n0 + 16]  = c11[r];
    }
}

// ---------------------------------------------------------------------------
// Post-processing: z=0 -> L2-normalize K per (row, head);
//                  z=1 -> L2-normalize V per (row, head);
//                  z=2 -> lr = sigmoid(lr_raw + b_lr).
// One wave handles one (row, head) chunk of 128 floats (4 per lane).
// ---------------------------------------------------------------------------
__global__ __launch_bounds__(256) void postproc(
    float* __restrict__ kbuf, float* __restrict__ vbuf,
    float* __restrict__ lrbuf, const float* __restrict__ b_lr)
{
    const int wid  = blockIdx.x * 8 + (threadIdx.x >> 5);
    const int lane = threadIdx.x & 31;
    const int n = wid >> 3;      // row in [0, B*T)
    const int h = wid & 7;       // head
    const size_t base = (size_t)n * DD + h * HD + lane * 4;

    if (blockIdx.z == 2) {
        float4 xv = *(const float4*)(lrbuf + base);
        float4 bv = *(const float4*)(b_lr + h * HD + lane * 4);
        xv.x = 1.0f / (1.0f + expf(-(xv.x + bv.x)));
        xv.y = 1.0f / (1.0f + expf(-(xv.y + bv.y)));
        xv.z = 1.0f / (1.0f + expf(-(xv.z + bv.z)));
        xv.w = 1.0f / (1.0f + expf(-(xv.w + bv.w)));
        *(float4*)(lrbuf + base) = xv;
    } else {
        float* buf = (blockIdx.z == 0) ? kbuf : vbuf;
        float4 xv = *(const float4*)(buf + base);
        float s = xv.x * xv.x + xv.y * xv.y + xv.z * xv.z + xv.w * xv.w;
        #pragma unroll
        for (int off = 16; off > 0; off >>= 1)
            s += __shfl_xor(s, off, 32);
        const float inv = 1.0f / fmaxf(sqrtf(s), 1e-12f);
        xv.x *= inv; xv.y *= inv; xv.z *= inv; xv.w *= inv;
        *(float4*)(buf + base) = xv;
    }
}

// ---------------------------------------------------------------------------
// Sequential delta-rule scan. One block per (b,h); 128 threads; thread i owns
// row i of the 128x128 f32 state entirely in VGPRs (32 float4). Per step:
//   v_old[i] = w[i]·k ; delta = lr[i]*(v[i]-v_old[i]);
//   w[i][:] += delta*k[:] ; o[i] = w[i]·q.
// All cross-thread data (k, q) is read as broadcast float4 loads (identical
// addresses across the wave -> single L0 request). No LDS, no barriers.
// ---------------------------------------------------------------------------
__global__ __launch_bounds__(128) void delta_scan(
    const float* __restrict__ qb, const float* __restrict__ kb,
    const float* __restrict__ vb, const float* __restrict__ lb,
    float* __restrict__ ob)
{
    const int b = blockIdx.x >> 3;
    const int h = blockIdx.x & 7;
    const int i = threadIdx.x;            // state row owned by this thread

    float4 w[32];
    #pragma unroll
    for (int j = 0; j < 32; ++j) w[j] = make_float4(0.f, 0.f, 0.f, 0.f);

    for (int t = 0; t < TT; ++t) {
        const size_t base = ((size_t)(b * TT + t)) * DD + h * HD;
        const float4* __restrict__ kp = (const float4*)(kb + base);
        const float4* __restrict__ qp = (const float4*)(qb + base);

        float4 kv[32];
        float vold = 0.f;
        #pragma unroll
        for (int j = 0; j < 32; ++j) {
            kv[j] = kp[j];
            vold += w[j].x * kv[j].x + w[j].y * kv[j].y
                  + w[j].z * kv[j].z + w[j].w * kv[j].w;
        }

        const float delta = lb[base + i] * (vb[base + i] - vold);

        float oi = 0.f;
        #pragma unroll
        for (int j = 0; j < 32; ++j) {
            w[j].x += delta * kv[j].x;
            w[j].y += delta * kv[j].y;
            w[j].z += delta * kv[j].z;
            w[j].w += delta * kv[j].w;
            float4 qv = qp[j];
            oi += w[j].x * qv.x + w[j].y * qv.y
                + w[j].z * qv.z + w[j].w * qv.w;
        }
        ob[base + i] = oi;
    }
}

// ---------------------------------------------------------------------------
// Host-side launch. Inputs: x, Wq, Wk, Wv, Wo, Wlr, b_lr (all f32).
// Workspace: q, k, v, lr, o buffers, each B*T*D f32 (5 x 32 MiB).
// ---------------------------------------------------------------------------
extern "C" void kernel_launch(void* const* d_in, const int* in_sizes, int n_in,
                              void* d_out, int out_size, void* d_ws, size_t ws_size,
                              hipStream_t stream)
{
    const float* x    = (const float*)d_in[0];
    const float* Wq   = (const float*)d_in[1];
    const float* Wk   = (const float*)d_in[2];
    const float* Wv   = (const float*)d_in[3];
    const float* Wo   = (const float*)d_in[4];
    const float* Wlr  = (const float*)d_in[5];
    const float* b_lr = (const float*)d_in[6];
    float* out = (float*)d_out;

    float* f = (float*)d_ws;
    const size_t SZ = (size_t)BB * TT * DD;   // 8,388,608 floats per buffer
    float* qb = f;
    float* kb = f + SZ;
    float* vb = f + 2 * SZ;
    float* lb = f + 3 * SZ;
    float* ob = f + 4 * SZ;

    // 1) Four fused projections: q, k_raw, v_raw, lr_raw.
    //    rows = B*T = 8192 -> 256 row-tiles of 32; cols = 1024 -> 4 blocks
    //    of 8 waves x 32 cols.
    dim3 gproj(DD / 256, (BB * TT) / 32, 4);
    gemm_xwT_f32<<<gproj, 256, 0, stream>>>(x, Wq, Wk, Wv, Wlr, qb, kb, vb, lb);

    // 2) Normalize k, v; sigmoid(lr + bias).
    dim3 gpost((BB * TT * HH) / 8, 1, 3);
    postproc<<<gpost, 256, 0, stream>>>(kb, vb, lb, b_lr);

    // 3) Sequential scan: 32 independent (b,h) recurrences.
    delta_scan<<<dim3(BB * HH), 128, 0, stream>>>(qb, kb, vb, lb, ob);

    // 4) Output projection: out = o @ Wo^T.
    dim3 gout(DD / 256, (BB * TT) / 32, 1);
    gemm_xwT_f32<<<gout, 256, 0, stream>>>(ob, Wo, Wo, Wo, Wo, out, out, out, out);
}